// SSPAttBlock_3195455668598
// MI455X (gfx1250) — compile-verified
//
#include <hip/hip_runtime.h>
#include <hip/hip_bf16.h>

// Problem constants (match reference)
#define BATCH 64
#define HH 256
#define WW 256
#define NPIX (HH * WW)          // 65536
#define SENT ((unsigned)NPIX)   // 65536
#define NUM_BINS 50

// TDM chunking (double-buffered 16 KB chunks)
#define CHUNK 4096              // floats per TDM chunk
#define NCHUNK (NPIX / CHUNK)   // 16

typedef unsigned int u32x4 __attribute__((ext_vector_type(4)));
typedef int          i32x4v __attribute__((ext_vector_type(4)));
typedef int          i32x8v __attribute__((ext_vector_type(8)));

#if defined(__HIP_DEVICE_COMPILE__) && __has_builtin(__builtin_amdgcn_tensor_load_to_lds) && __has_builtin(__builtin_amdgcn_s_wait_tensorcnt)
#define HAS_TDM_LD 1
#else
#define HAS_TDM_LD 0
#endif
#if defined(__HIP_DEVICE_COMPILE__) && __has_builtin(__builtin_amdgcn_tensor_store_from_lds) && __has_builtin(__builtin_amdgcn_s_wait_tensorcnt)
#define HAS_TDM_ST 1
#else
#define HAS_TDM_ST 0
#endif

#if HAS_TDM_LD || HAS_TDM_ST
// Build 1-D D# (group0 in g0, group1 in g1): nelem f32 elements, LDS byte
// offset lds_off, global byte address gaddr. Groups 2/3 unused (zero).
__device__ __forceinline__ void tdm_build_1d(u32x4& g0, i32x8v& g1,
                                             unsigned lds_off,
                                             unsigned long long gaddr,
                                             unsigned nelem) {
  g0.x = 1u;                                             // count=1 (valid), user mode
  g0.y = lds_off;                                        // lds_addr (bytes)
  g0.z = (unsigned)(gaddr & 0xFFFFFFFFull);              // global_addr[31:0]
  g0.w = ((unsigned)(gaddr >> 32) & 0x01FFFFFFu)         // global_addr[56:32]
         | 0x80000000u;                                  // type=2 ("image")
  g1[0] = (int)(2u << 16);                               // data_size=2 -> 4 bytes
  g1[1] = (int)((nelem & 0xFFFFu) << 16);                // tensor_dim0[15:0] @ bits63:48
  g1[2] = (int)(((nelem >> 16) & 0xFFFFu)                // tensor_dim0[31:16]
                | (1u << 16));                           // tensor_dim1 = 1
  g1[3] = (int)((nelem & 0xFFFFu) << 16);                // tile_dim0 @ bits127:112
  g1[4] = 0;                                             // tile_dim1=0, tile_dim2=0 (1-D)
  g1[5] = (int)nelem;                                    // tensor_dim0_stride[31:0]
  g1[6] = 0;
  g1[7] = 0;
}
#endif

#if HAS_TDM_LD
__device__ __forceinline__ void tdm_load_1d(unsigned lds_off,
                                            unsigned long long gaddr,
                                            unsigned nelem) {
  u32x4 g0; i32x8v g1;
  tdm_build_1d(g0, g1, lds_off, gaddr, nelem);
  i32x4v z4 = {0, 0, 0, 0};
#if defined(__clang_major__) && (__clang_major__ >= 23)
  i32x8v z8 = {0, 0, 0, 0, 0, 0, 0, 0};
  __builtin_amdgcn_tensor_load_to_lds(g0, g1, z4, z4, z8, 0);
#else
  __builtin_amdgcn_tensor_load_to_lds(g0, g1, z4, z4, 0);
#endif
}
#endif

#if HAS_TDM_ST
__device__ __forceinline__ void tdm_store_1d(unsigned lds_off,
                                             unsigned long long gaddr,
                                             unsigned nelem) {
  u32x4 g0; i32x8v g1;
  tdm_build_1d(g0, g1, lds_off, gaddr, nelem);
  i32x4v z4 = {0, 0, 0, 0};
#if defined(__clang_major__) && (__clang_major__ >= 23)
  i32x8v z8 = {0, 0, 0, 0, 0, 0, 0, 0};
  __builtin_amdgcn_tensor_store_from_lds(g0, g1, z4, z4, z8, 0);
#else
  __builtin_amdgcn_tensor_store_from_lds(g0, g1, z4, z4, 0);
#endif
}
#endif

__device__ __forceinline__ unsigned lds_offset_of(const void* p) {
  return (unsigned)(unsigned long long)p;   // low 32 bits of flat LDS addr = offset
}

// ---------------------------------------------------------------------------
// Kernel 1: per-batch 50-bin histogram -> threshold  thr[b] = ind_sec / 50
// One block per batch image. TDM streams the image into LDS double buffers.
// ---------------------------------------------------------------------------
__global__ __launch_bounds__(256) void ssp_hist_thresh(
    const float* __restrict__ att, float* __restrict__ thr) {
  __shared__ float stage[2][CHUNK];   // 32 KB staging
  __shared__ unsigned hist[NUM_BINS];

  const int tid = threadIdx.x;
  const int b = blockIdx.x;
  const unsigned long long base =
      (unsigned long long)(const void*)(att + (size_t)b * NPIX);

  for (int i = tid; i < NUM_BINS; i += 256) hist[i] = 0u;

#if HAS_TDM_LD
  const unsigned lds0 = lds_offset_of(&stage[0][0]);
  const unsigned lds1 = lds_offset_of(&stage[1][0]);
  const bool issuer = (tid < 32);     // wave 0 issues the DMA (once per wave)

  if (issuer) tdm_load_1d(lds0, base, CHUNK);
  for (int c = 0; c < NCHUNK; ++c) {
    if (c + 1 < NCHUNK) {
      if (issuer)
        tdm_load_1d(((c + 1) & 1) ? lds1 : lds0,
                    base + (unsigned long long)(c + 1) * CHUNK * 4ull, CHUNK);
      __builtin_amdgcn_s_wait_tensorcnt((short)1);  // chunk c complete
    } else {
      __builtin_amdgcn_s_wait_tensorcnt((short)0);
    }
    __syncthreads();
    const float* buf = stage[c & 1];
#pragma unroll
    for (int i = 0; i < CHUNK / 256; ++i) {
      float v = buf[i * 256 + tid];
      int idx = (int)(v * (float)NUM_BINS);
      idx = idx < 0 ? 0 : (idx > NUM_BINS - 1 ? NUM_BINS - 1 : idx);
      atomicAdd(&hist[idx], 1u);
    }
    __syncthreads();                  // done reading buf before it is refilled
  }
#else
  __syncthreads();
  const float* ab = att + (size_t)b * NPIX;
  for (int i = tid; i < NPIX; i += 256) {
    float v = ab[i];
    int idx = (int)(v * (float)NUM_BINS);
    idx = idx < 0 ? 0 : (idx > NUM_BINS - 1 ? NUM_BINS - 1 : idx);
    atomicAdd(&hist[idx], 1u);
  }
  __syncthreads();
#endif

  __syncthreads();
  if (tid == 0) {
    int best = 0;
    unsigned bc = hist[0];
    for (int i = 1; i < NUM_BINS; ++i)
      if (hist[i] > bc) { bc = hist[i]; best = i; }   // first argmax
    int sec = 0;
    long long sc = -2;
    for (int i = 0; i < NUM_BINS; ++i) {
      long long v = (i > best) ? (long long)hist[i] : -1ll;
      if (v > sc) { sc = v; sec = i; }                // first argmax of masked
    }
    thr[b] = (float)sec / (float)NUM_BINS;
  }
}

// ---------------------------------------------------------------------------
// Kernel 2: per-batch connected components fully in LDS (256 KB label image,
// CDNA5 320 KB/WGP). TDM streams att in for label init, and streams the final
// result out via tensor_store_from_lds. LDS reused as area-count table.
// Grid: BATCH blocks x 1024 threads; dynamic LDS = (NPIX+32+2*CHUNK)*4 bytes.
// ---------------------------------------------------------------------------
__global__ __launch_bounds__(1024) void ssp_cc_apply(
    const float* __restrict__ att, const float* __restrict__ thr,
    unsigned* __restrict__ wslab, float* __restrict__ out) {
  extern __shared__ unsigned smem[];
  unsigned* lab = smem;                         // [0,NPIX) labels; [NPIX] flag
  float* stage = (float*)(smem + NPIX + 32);    // 2*CHUNK floats staging

  const int tid = threadIdx.x;
  const int b = blockIdx.x;
  const float t = thr[b];
  const float* a = att + (size_t)b * NPIX;
  unsigned* wl = wslab + (size_t)b * NPIX;
  float* o = out + (size_t)b * NPIX;
  const int PPT = NPIX / 1024;                  // 64 pixels per thread

  // ---- Phase 1: init labels (fg = att > thr), att streamed in via TDM ----
#if HAS_TDM_LD
  {
    const unsigned lds0 = lds_offset_of(&stage[0]);
    const unsigned lds1 = lds_offset_of(&stage[CHUNK]);
    const unsigned long long base = (unsigned long long)(const void*)a;
    const bool issuer = (tid < 32);
    if (issuer) tdm_load_1d(lds0, base, CHUNK);
    for (int c = 0; c < NCHUNK; ++c) {
      if (c + 1 < NCHUNK) {
        if (issuer)
          tdm_load_1d(((c + 1) & 1) ? lds1 : lds0,
                      base + (unsigned long long)(c + 1) * CHUNK * 4ull, CHUNK);
        __builtin_amdgcn_s_wait_tensorcnt((short)1);
      } else {
        __builtin_amdgcn_s_wait_tensorcnt((short)0);
      }
      __syncthreads();
      const float* buf = &stage[(c & 1) * CHUNK];
#pragma unroll
      for (int j = 0; j < CHUNK / 1024; ++j) {
        int p = c * CHUNK + j * 1024 + tid;
        float v = buf[j * 1024 + tid];
        lab[p] = (v > t) ? (unsigned)p : SENT;
      }
      __syncthreads();
    }
  }
#else
  for (int i = 0; i < PPT; ++i) {
    int p = i * 1024 + tid;
    float v = a[p];
    lab[p] = (v > t) ? (unsigned)p : SENT;
  }
  __syncthreads();
#endif

  // ---- Phase 2: monotone min-propagation + pointer chasing to fixed point --
  for (;;) {
    if (tid == 0) lab[NPIX] = 0u;
    __syncthreads();
    unsigned changed = 0u;
    for (int i = 0; i < PPT; ++i) {
      int p = i * 1024 + tid;
      unsigned l = lab[p];
      if (l == SENT) continue;
      unsigned m = l;
      int r = p >> 8, c = p & 255;
      if (r > 0) { unsigned n = lab[p - 256]; m = n < m ? n : m; }
      if (r < HH - 1) { unsigned n = lab[p + 256]; m = n < m ? n : m; }
      if (c > 0) { unsigned n = lab[p - 1]; m = n < m ? n : m; }
      if (c < WW - 1) { unsigned n = lab[p + 1]; m = n < m ? n : m; }
      for (;;) {                       // chase toward root (monotone decreasing)
        unsigned n = lab[m];
        if (n >= m) break;
        m = n;
      }
      if (m < l) { lab[p] = m; changed = 1u; }
    }
    if (changed) lab[NPIX] = 1u;       // benign same-value race
    __syncthreads();
    if (lab[NPIX] == 0u) break;
    __syncthreads();
  }

  // ---- Phase 3: spill labels, reuse LDS labels as per-root area counters ---
  for (int i = 0; i < PPT; ++i) {
    int p = i * 1024 + tid;
    wl[p] = lab[p];
  }
  __syncthreads();
  for (int i = 0; i < PPT; ++i) lab[i * 1024 + tid] = 0u;
  __syncthreads();
  for (int i = 0; i < PPT; ++i) {
    int p = i * 1024 + tid;
    unsigned l = wl[p];                // read own write, program order
    if (l != SENT) atomicAdd(&lab[l], 1u);
  }
  __syncthreads();

  // ---- Phase 4: output = fg ? att**(1/sqrt(area)) : att, TDM store out -----
#if HAS_TDM_ST
  {
    const unsigned long long obase = (unsigned long long)(const void*)o;
    const bool issuer = (tid < 32);
    for (int c = 0; c < NCHUNK; ++c) {
      // Buffer (c&1) is free once stores through chunk c-2 completed.
      __builtin_amdgcn_s_wait_tensorcnt((short)1);
      __syncthreads();
      float* obuf = &stage[(c & 1) * CHUNK];
#pragma unroll
      for (int j = 0; j < CHUNK / 1024; ++j) {
        int p = c * CHUNK + j * 1024 + tid;
        float v = a[p];
        unsigned l = wl[p];
        float r = v;
        if (l != SENT) {
          unsigned area = lab[l];
          float e = 1.0f / sqrtf((float)(area < 1u ? 1u : area));
          r = powf(v, e);
        }
        obuf[j * 1024 + tid] = r;
      }
      __syncthreads();                 // chunk staged before DMA reads it
      if (issuer)
        tdm_store_1d(lds_offset_of(obuf),
                     obase + (unsigned long long)c * CHUNK * 4ull, CHUNK);
    }
    __builtin_amdgcn_s_wait_tensorcnt((short)0);   // drain before exit
  }
#else
  for (int i = 0; i < PPT; ++i) {
    int p = i * 1024 + tid;
    float v = a[p];
    unsigned l = wl[p];
    float r = v;
    if (l != SENT) {
      unsigned area = lab[l];
      float e = 1.0f / sqrtf((float)(area < 1u ? 1u : area));
      r = powf(v, e);
    }
    o[p] = r;
  }
#endif
}

// ---------------------------------------------------------------------------
extern "C" void kernel_launch(void* const* d_in, const int* in_sizes, int n_in,
                              void* d_out, int out_size, void* d_ws,
                              size_t ws_size, hipStream_t stream) {
  (void)in_sizes; (void)n_in; (void)out_size; (void)ws_size;
  const float* att = (const float*)d_in[0];
  float* out = (float*)d_out;

  unsigned char* ws = (unsigned char*)d_ws;
  unsigned* wslab = (unsigned*)ws;                         // BATCH*NPIX u32
  float* thr = (float*)(ws + (size_t)BATCH * NPIX * 4);    // BATCH f32

  const size_t shmem =
      ((size_t)NPIX + 32 + 2 * CHUNK) * sizeof(unsigned);  // ~288 KB
  (void)hipFuncSetAttribute((const void*)ssp_cc_apply,
                            hipFuncAttributeMaxDynamicSharedMemorySize,
                            (int)shmem);

  ssp_hist_thresh<<<BATCH, 256, 0, stream>>>(att, thr);
  ssp_cc_apply<<<BATCH, 1024, shmem, stream>>>(att, thr, wslab, out);
}